// GCN_44160853737649
// MI455X (gfx1250) — compile-verified
//
#include <hip/hip_runtime.h>
#include <hip/hip_bf16.h>

// ---------------------------------------------------------------------------
// GCN 3-layer forward for MI455X (gfx1250, wave32).
//
// Roofline: GEMMs ~5 GFLOP (negligible); edge gather + scatter-add ~2.5 GB of
// irregular traffic dominates (~120us at 23.3 TB/s). Intermediates (<=51MB)
// fit in the 192MB L2, so a staged multi-kernel pipeline is bandwidth-optimal.
//
// GEMM: native fp32 WMMA (v_wmma_f32_16x16x4_f32), one 16x16 C tile per wave.
// Each wave pins its column-tile's entire B panel (Fin x 16 of W) in VGPRs
// (<=64 VGPRs) and sweeps row tiles, so the inner loop is just
// global_load_b64 (A frag) + v_wmma — no LDS, no per-iteration index math.
// ---------------------------------------------------------------------------

typedef __attribute__((ext_vector_type(2))) float v2f;
typedef __attribute__((ext_vector_type(8))) float v8f;

#define EPS 1e-5f
#define NEG_SLOPE 0.01f

// ---------------------------------------------------------------------------
__global__ __launch_bounds__(256) void zero_kernel(float* __restrict__ p, long long n) {
    long long stride = (long long)gridDim.x * blockDim.x;
    for (long long i = (long long)blockIdx.x * blockDim.x + threadIdx.x; i < n; i += stride)
        p[i] = 0.0f;
}

// deg[dst[e]] += 1  (self-loop folded in later as +1)
__global__ __launch_bounds__(256) void deg_kernel(const long long* __restrict__ dst,
                                                  float* __restrict__ deg, long long E) {
    long long stride = (long long)gridDim.x * blockDim.x;
    for (long long e = (long long)blockIdx.x * blockDim.x + threadIdx.x; e < E; e += stride) {
        __hip_atomic_fetch_add(&deg[dst[e]], 1.0f, __ATOMIC_RELAXED, __HIP_MEMORY_SCOPE_AGENT);
    }
}

// dinv[i] = rsqrt(deg[i] + 1)   (+1 = self-loop; deg+1 > 0 always)
__global__ __launch_bounds__(256) void dinv_kernel(float* __restrict__ deg, int n) {
    long long stride = (long long)gridDim.x * blockDim.x;
    for (long long i = (long long)blockIdx.x * blockDim.x + threadIdx.x; i < n; i += stride) {
        float d = deg[i] + 1.0f;
        deg[i] = (d > 0.0f) ? __frsqrt_rn(d) : 0.0f;
    }
}

// norm[e] = dinv[src[e]] * dinv[dst[e]]  (shared by all three layers)
__global__ __launch_bounds__(256) void norm_kernel(const long long* __restrict__ src,
                                                   const long long* __restrict__ dst,
                                                   const float* __restrict__ dinv,
                                                   float* __restrict__ norm, long long E) {
    long long stride = (long long)gridDim.x * blockDim.x;
    for (long long e = (long long)blockIdx.x * blockDim.x + threadIdx.x; e < E; e += stride)
        norm[e] = dinv[src[e]] * dinv[dst[e]];
}

// ---------------------------------------------------------------------------
// H = X @ W with v_wmma_f32_16x16x4_f32, B panel resident in VGPRs.
//   X: nrows x Fin, W: Fin x Fout (row-major), H: nrows x Fout.
// Fout16 = ceil16(Fout); Fout=8 layer uses one zero-padded 16-wide tile with
// a column-guarded store.
//
// Fragment layouts per CDNA5 ISA 7.12.2 (32-bit A 16x4 / C 16x16):
//   A: lane m=lane&15 is row M; vgpr{0,1} hold K = k + (lane>>4)*2 + {0,1}
//   C: vgpr v holds row M = v + (lane>>4)*8, column N = lane&15
// B (4x16) mirrors A: lane&15 = column N; vgpr{0,1} = K rows (lane>>4)*2+{0,1}
// ---------------------------------------------------------------------------
template <int Fin, int Fout, int Fout16>
__global__ __launch_bounds__(256) void gemm_wmma_f32(const float* __restrict__ X,
                                                     const float* __restrict__ Wm,
                                                     float* __restrict__ H, int nrows) {
    constexpr int KSTEPS = Fin / 4;      // WMMAs per tile
    constexpr int TPR    = Fout16 / 16;  // column tiles (8 / 4 / 1)

    const int lane = threadIdx.x & 31;
    const int wave = threadIdx.x >> 5;
    const int wavesPerBlock = blockDim.x >> 5;  // 8
    const int m   = lane & 15;                  // row (A) / col (B,C) in tile
    const int khi = (lane >> 4) * 2;            // K sub-offset, upper half-wave

    const int ct  = wave % TPR;
    const int col = ct * 16 + m;

    // --- Pin B panel (Fin x 16 slice of W) for this column tile in VGPRs ---
    v2f bfrag[KSTEPS];
#pragma unroll
    for (int j = 0; j < KSTEPS; ++j) {
        int k = j * 4 + khi;
        if (col < Fout) {
            bfrag[j].x = Wm[(k + 0) * Fout + col];
            bfrag[j].y = Wm[(k + 1) * Fout + col];
        } else {
            bfrag[j].x = 0.0f;
            bfrag[j].y = 0.0f;
        }
    }

    // --- Sweep row tiles; inner loop = A b64 load + wmma only ---
    const int rowTiles  = (nrows + 15) >> 4;
    const int rtGroups  = wavesPerBlock / TPR;  // row-tile groups per block
    const int rtInBlock = wave / TPR;
    const int rtStride  = gridDim.x * rtGroups;

    for (int rt = blockIdx.x * rtGroups + rtInBlock; rt < rowTiles; rt += rtStride) {
        int rowTile = rt << 4;
        int rowA = rowTile + m;
        if (rowA >= nrows) rowA = nrows - 1;  // safe clamp (nrows%16==0 here)
        const float* xrow = X + (size_t)rowA * Fin + khi;

        v8f acc = {};
#pragma unroll
        for (int j = 0; j < KSTEPS; ++j) {
            v2f a = *(const v2f*)(xrow + j * 4);
            acc = __builtin_amdgcn_wmma_f32_16x16x4_f32(
                /*neg_a=*/false, a, /*neg_b=*/false, bfrag[j],
                /*c_mod=*/(short)0, acc, /*reuse_a=*/false, /*reuse_b=*/false);
        }

        if (col < Fout) {
            int rbase = rowTile + ((lane >> 4) << 3);
#pragma unroll
            for (int v = 0; v < 8; ++v) {
                int r = rbase + v;
                if (r < nrows) H[(size_t)r * Fout + col] = acc[v];
            }
        }
    }
}

// ---------------------------------------------------------------------------
// out[dst[e], :] += norm[e] * H[src[e], :]   (float4 gather, fp32 atomic adds)
// logFv = log2(F/4); F in {128,64,8}.
// ---------------------------------------------------------------------------
__global__ __launch_bounds__(256) void agg_kernel(const float* __restrict__ H,
                                                  const long long* __restrict__ src,
                                                  const long long* __restrict__ dst,
                                                  const float* __restrict__ norm,
                                                  float* __restrict__ out,
                                                  long long E, int F, int logFv) {
    const int Fv = 1 << logFv;
    const long long total = E << logFv;
    const long long stride = (long long)gridDim.x * blockDim.x;
    for (long long t = (long long)blockIdx.x * blockDim.x + threadIdx.x; t < total; t += stride) {
        long long e = t >> logFv;
        int f = (int)(t & (Fv - 1)) << 2;
        long long s = src[e], d = dst[e];
        float w = norm[e];
        float4 h = *(const float4*)(H + (size_t)s * F + f);
        float* op = out + (size_t)d * F + f;
        __hip_atomic_fetch_add(op + 0, w * h.x, __ATOMIC_RELAXED, __HIP_MEMORY_SCOPE_AGENT);
        __hip_atomic_fetch_add(op + 1, w * h.y, __ATOMIC_RELAXED, __HIP_MEMORY_SCOPE_AGENT);
        __hip_atomic_fetch_add(op + 2, w * h.z, __ATOMIC_RELAXED, __HIP_MEMORY_SCOPE_AGENT);
        __hip_atomic_fetch_add(op + 3, w * h.w, __ATOMIC_RELAXED, __HIP_MEMORY_SCOPE_AGENT);
    }
}

// out = leaky_relu(agg + dinv[i]^2 * H[i,:] + bias)   (self-loop + bias fused)
__global__ __launch_bounds__(256) void post_kernel(const float* __restrict__ agg,
                                                   const float* __restrict__ H,
                                                   const float* __restrict__ dinv,
                                                   const float* __restrict__ bias,
                                                   float* __restrict__ out,
                                                   int nrows, int F, int logF) {
    const long long total = (long long)nrows * F;
    const long long stride = (long long)gridDim.x * blockDim.x;
    for (long long t = (long long)blockIdx.x * blockDim.x + threadIdx.x; t < total; t += stride) {
        long long i = t >> logF;
        int f = (int)(t & (F - 1));
        float di = dinv[i];
        float v = agg[t] + di * di * H[t] + bias[f];
        out[t] = (v > 0.0f) ? v : NEG_SLOPE * v;
    }
}

// Column sums / sums-of-squares for BatchNorm (training-mode batch stats).
__global__ __launch_bounds__(256) void bn_stats(const float* __restrict__ X,
                                                float* __restrict__ sums,
                                                float* __restrict__ sqs,
                                                int nrows, int F, int logF) {
    int c = threadIdx.x & (F - 1);
    int rpb = blockDim.x >> logF;                  // rows handled per block pass
    int r = blockIdx.x * rpb + (threadIdx.x >> logF);
    int rstride = gridDim.x * rpb;
    float s = 0.0f, q = 0.0f;
    for (; r < nrows; r += rstride) {
        float v = X[(size_t)r * F + c];
        s += v;
        q += v * v;
    }
    __hip_atomic_fetch_add(&sums[c], s, __ATOMIC_RELAXED, __HIP_MEMORY_SCOPE_AGENT);
    __hip_atomic_fetch_add(&sqs[c],  q, __ATOMIC_RELAXED, __HIP_MEMORY_SCOPE_AGENT);
}

// x = (x - mean) * rsqrt(var + eps) * gamma + beta   (biased variance)
__global__ __launch_bounds__(256) void bn_apply(float* __restrict__ X,
                                                const float* __restrict__ sums,
                                                const float* __restrict__ sqs,
                                                const float* __restrict__ gamma,
                                                const float* __restrict__ beta,
                                                int nrows, int F, int logF) {
    const float invN = 1.0f / (float)nrows;
    const long long total = (long long)nrows * F;
    const long long stride = (long long)gridDim.x * blockDim.x;
    for (long long t = (long long)blockIdx.x * blockDim.x + threadIdx.x; t < total; t += stride) {
        int f = (int)(t & (F - 1));
        float m = sums[f] * invN;
        float var = sqs[f] * invN - m * m;
        X[t] = (X[t] - m) * __frsqrt_rn(var + EPS) * gamma[f] + beta[f];
    }
}

// ---------------------------------------------------------------------------
static inline size_t align256(size_t x) { return (x + 255) & ~(size_t)255; }

extern "C" void kernel_launch(void* const* d_in, const int* in_sizes, int n_in,
                              void* d_out, int out_size, void* d_ws, size_t ws_size,
                              hipStream_t stream) {
    const float*     x   = (const float*)d_in[0];
    const long long* ei  = (const long long*)d_in[1];   // int64 [2, E]
    // d_in[2] = edge_weights: unused by the reference forward
    const float* W1 = (const float*)d_in[3];  const float* b1 = (const float*)d_in[4];
    const float* W2 = (const float*)d_in[5];  const float* b2 = (const float*)d_in[6];
    const float* W3 = (const float*)d_in[7];  const float* b3 = (const float*)d_in[8];
    const float* g1 = (const float*)d_in[9];  const float* be1 = (const float*)d_in[10];
    const float* g2 = (const float*)d_in[11]; const float* be2 = (const float*)d_in[12];

    const long long E = (long long)in_sizes[1] / 2;
    const int N = in_sizes[0] / 128;
    const long long* src = ei;
    const long long* dst = ei + E;

    // Workspace carve-up
    char* base = (char*)d_ws;
    size_t off = 0;
    float* dinv = (float*)(base + off); off = align256(off + (size_t)N * 4);
    float* norm = (float*)(base + off); off = align256(off + (size_t)E * 4);
    float* A    = (float*)(base + off); off = align256(off + (size_t)N * 128 * 4);  // H (pre-agg)
    float* B    = (float*)(base + off); off = align256(off + (size_t)N * 128 * 4);  // agg/act
    float* C    = (float*)(base + off); off = align256(off + (size_t)N * 8 * 4);    // layer-3 agg
    float* bnS  = (float*)(base + off); off = align256(off + 256 * 4);              // sums(128)+sqs(128)
    float* bnQ  = bnS + 128;
    (void)ws_size; (void)n_in; (void)out_size;

    const int T = 256;
    const int gZeroBig = 16384, gEdge = 8192, gElem = 8192, gStat = 1024, gGemm = 512;

    // --- degree / normalization (unit edge weights, in-degree on dst) ---
    zero_kernel<<<(N + T - 1) / T, T, 0, stream>>>(dinv, N);
    deg_kernel<<<gEdge, T, 0, stream>>>(dst, dinv, E);
    dinv_kernel<<<(N + T - 1) / T, T, 0, stream>>>(dinv, N);
    norm_kernel<<<gEdge, T, 0, stream>>>(src, dst, dinv, norm, E);

    // ---------------- Layer 1: 128 -> 128 ----------------
    {
        gemm_wmma_f32<128, 128, 128><<<gGemm, T, 0, stream>>>(x, W1, A, N);
        zero_kernel<<<gZeroBig, T, 0, stream>>>(B, (long long)N * 128);
        agg_kernel<<<gEdge, T, 0, stream>>>(A, src, dst, norm, B, E, 128, 5);
        post_kernel<<<gElem, T, 0, stream>>>(B, A, dinv, b1, B, N, 128, 7);
        zero_kernel<<<1, T, 0, stream>>>(bnS, 256);
        bn_stats<<<gStat, T, 0, stream>>>(B, bnS, bnQ, N, 128, 7);
        bn_apply<<<gElem, T, 0, stream>>>(B, bnS, bnQ, g1, be1, N, 128, 7);
    }

    // ---------------- Layer 2: 128 -> 64 ----------------
    {
        gemm_wmma_f32<128, 64, 64><<<gGemm, T, 0, stream>>>(B, W2, A, N);
        zero_kernel<<<gZeroBig, T, 0, stream>>>(B, (long long)N * 64);
        agg_kernel<<<gEdge, T, 0, stream>>>(A, src, dst, norm, B, E, 64, 4);
        post_kernel<<<gElem, T, 0, stream>>>(B, A, dinv, b2, B, N, 64, 6);
        zero_kernel<<<1, T, 0, stream>>>(bnS, 256);
        bn_stats<<<gStat, T, 0, stream>>>(B, bnS, bnQ, N, 64, 6);
        bn_apply<<<gElem, T, 0, stream>>>(B, bnS, bnQ, g2, be2, N, 64, 6);
    }

    // ---------------- Layer 3: 64 -> 8 (padded tile, no BN) ----------------
    {
        gemm_wmma_f32<64, 8, 16><<<gGemm, T, 0, stream>>>(B, W3, A, N);
        zero_kernel<<<gZeroBig, T, 0, stream>>>(C, (long long)N * 8);
        agg_kernel<<<gEdge, T, 0, stream>>>(A, src, dst, norm, C, E, 8, 1);
        post_kernel<<<gElem, T, 0, stream>>>(C, A, dinv, b3, (float*)d_out, N, 8, 3);
    }
}